// RelativePositionalMask_38482906972941
// MI455X (gfx1250) — compile-verified
//
#include <hip/hip_runtime.h>
#include <hip/hip_bf16.h>

#define N 3072
#define NHEAD 8
#define NN (N * N)  // 9437184, fits in 32-bit

typedef __attribute__((ext_vector_type(2))) float v2f;
typedef __attribute__((ext_vector_type(4))) float v4f;
typedef __attribute__((ext_vector_type(8))) float v8f;

// out[h, j, i] = bias[idx(i,j), h]
// idx = sidx + 32*tidx
// sidx = clamp(ceil(31/(2*log2(257)) * log2(d^2(i,j))), 0, 31)
// tidx = clamp(ceil((f_i - f_j)/2) + 16, 0, 32)
__global__ __launch_bounds__(256) void RelativePositionalMask_kernel(
    const float* __restrict__ coords,  // (N,3): frame, x, y
    const float* __restrict__ bias,    // (1056, 8)
    float* __restrict__ out)           // (8, N, N)
{
    const int lane = threadIdx.x & 31;
    const int wave = threadIdx.x >> 5;
    const int half = lane >> 4;   // which 16-lane half of the wave
    const int l16  = lane & 15;

    const int ibase = blockIdx.x * 128 + wave * 16;  // 24 blocks * 128 = 3072
    const int jbase = blockIdx.y * 16;               // 192 blocks * 16 = 3072

    // Each lane loads the j-point for row m=l16 and the i-point for col n=l16.
    const float fj = coords[(jbase + l16) * 3 + 0];
    const float xj = coords[(jbase + l16) * 3 + 1];
    const float yj = coords[(jbase + l16) * 3 + 2];
    const float fi = coords[(ibase + l16) * 3 + 0];
    const float xi = coords[(ibase + l16) * 3 + 1];
    const float yi = coords[(ibase + l16) * 3 + 2];

    const float nj = xj * xj + yj * yj;  // |p_j|^2
    const float ni = xi * xi + yi * yi;  // |p_i|^2

    // A (16x4 f32): lanes 0-15 hold K=0,1 ; lanes 16-31 hold K=2,3 (M = lane&15)
    // B (4x16 f32): same split over K, N = lane&15
    // Spatial:  A[m] = [xj, yj, |pj|^2, 1],  B[:,n] = [-2xi, -2yi, 1, |pi|^2]
    //   -> D[m][n] = |pj|^2 + |pi|^2 - 2(xj*xi + yj*yi) = d^2(j,i)
    // Temporal: A[m] = [1, fj, 0, 0],        B[:,n] = [fi, -1, 0, 0]
    //   -> D[m][n] = fi - fj
    v2f a_sp, b_sp, a_tm, b_tm;
    a_sp.x = (half == 0) ? xj : nj;
    a_sp.y = (half == 0) ? yj : 1.0f;
    b_sp.x = (half == 0) ? (-2.0f * xi) : 1.0f;
    b_sp.y = (half == 0) ? (-2.0f * yi) : ni;
    a_tm.x = (half == 0) ? 1.0f : 0.0f;
    a_tm.y = (half == 0) ? fj   : 0.0f;
    b_tm.x = (half == 0) ? fi   : 0.0f;
    b_tm.y = (half == 0) ? -1.0f : 0.0f;

    v8f czero = {};
    v8f d2 = __builtin_amdgcn_wmma_f32_16x16x4_f32(
        false, a_sp, false, b_sp, (short)0, czero, false, false);
    v8f td = __builtin_amdgcn_wmma_f32_16x16x4_f32(
        false, a_tm, false, b_tm, (short)0, czero, false, false);

    // C/D layout: element e of this lane is (M = e + 8*half, N = l16)
    const int m0 = half * 8;
    // 31 / (2*log2(257)) = 31 / 16.0112497 ...
    const float KSP = 1.9361387f;

#pragma unroll
    for (int e = 0; e < 8; ++e) {
        // spatial bin: ceil(KSP * log2(d^2)); d^2==0 -> -inf -> clamps to 0
        float d2v = fmaxf(d2[e], 0.0f);
        float sf  = ceilf(KSP * __log2f(d2v));
        int sidx  = (int)fminf(fmaxf(sf, 0.0f), 31.0f);

        // temporal bin: ceil(t/2) + 16, clamp [0,32]
        float tf  = ceilf(td[e] * 0.5f) + 16.0f;
        int tidx  = (int)fminf(fmaxf(tf, 0.0f), 32.0f);

        int idx = (tidx << 5) + sidx;

        // gather 8 heads (32B, cache-resident table)
        const v4f* bp = (const v4f*)(bias + idx * NHEAD);
        v4f b0 = bp[0];
        v4f b1 = bp[1];
        float vals[8] = {b0.x, b0.y, b0.z, b0.w, b1.x, b1.y, b1.z, b1.w};

        unsigned off = (unsigned)((jbase + m0 + e) * N + ibase + l16);
#pragma unroll
        for (int h = 0; h < 8; ++h) {
            out[off + (unsigned)h * (unsigned)NN] = vals[h];
        }
    }
}

extern "C" void kernel_launch(void* const* d_in, const int* in_sizes, int n_in,
                              void* d_out, int out_size, void* d_ws, size_t ws_size,
                              hipStream_t stream) {
    const float* coords = (const float*)d_in[0];  // (N,3) f32
    const float* bias   = (const float*)d_in[1];  // (1056,8) f32
    // d_in[2]/d_in[3] (bin edges) are folded into closed-form index math.
    float* out = (float*)d_out;                   // (8,N,N) f32

    dim3 grid(24, 192);   // i tiles: 24*8 waves*16 = 3072 ; j tiles: 192*16 = 3072
    dim3 block(256);      // 8 wave32 -> 8 independent 16x16 tiles per block
    RelativePositionalMask_kernel<<<grid, block, 0, stream>>>(coords, bias, out);
}